// TNet_20968030339253
// MI455X (gfx1250) — compile-verified
//
#include <hip/hip_runtime.h>

typedef __attribute__((ext_vector_type(2))) float v2f;
typedef __attribute__((ext_vector_type(8))) float v8f;

namespace {
constexpr int kB = 32;
constexpr int kH = 384;
constexpr int kW = 384;
constexpr int kPitch = 392;   // floats per LDS row (data at [4..387], halo at 3 / 388)
constexpr int kRows  = 4;     // rows y-1, y, y+1 + one all-zero pad row (taps 9..11)
}

// Fold: channel-collapse of conv1, BN scale/shift, 1x1 conv weights.
// ws[0..35]  = scale[o]*sum_i conv1_w[o,i,ky,kx]   (o*9 + ky*3+kx)
// ws[36..39] = beta[o] - scale[o]*mean[o]
// ws[40..43] = conv2_w[o], o=0..3
// ws[44]     = conv2_w[4]+conv2_w[5]+conv2_w[6]+conv2_w[7]
// ws[45]     = conv2_b[0]
__global__ void prep_kernel(const float* __restrict__ w1,
                            const float* __restrict__ gamma,
                            const float* __restrict__ beta,
                            const float* __restrict__ mean,
                            const float* __restrict__ var,
                            const float* __restrict__ w2,
                            const float* __restrict__ b2,
                            float* __restrict__ ws) {
  int o = threadIdx.x;
  if (o < 4) {
    float sc = gamma[o] * rsqrtf(var[o] + 1e-5f);
    for (int t = 0; t < 9; ++t) {
      float s = 0.f;
      for (int i = 0; i < 4; ++i) s += w1[o * 36 + i * 9 + t];
      ws[o * 9 + t] = sc * s;
    }
    ws[36 + o] = beta[o] - sc * mean[o];
    ws[40 + o] = w2[o];
  } else if (o == 4) {
    ws[44] = w2[4] + w2[5] + w2[6] + w2[7];
    ws[45] = b2[0];
  }
}

// One block (8 waves) = one output row (b, y). Rows y-1..y+1 staged into LDS
// via global_load_async_to_lds_b128 with zero halos materialized; each wave
// computes 48 px as 3 segments of 16 via V_WMMA_F32_16X16X4_F32
// (M=channel 0..3, N=16 pixels, K=3x3 taps padded to 12).
__global__ __launch_bounds__(256) void tnet_wmma_kernel(
    const float* __restrict__ depth, const float* __restrict__ ws,
    float* __restrict__ out) {
  __shared__ float smem[kRows * kPitch];

  const int tid  = threadIdx.x;
  const int lane = tid & 31;
  const int wid  = tid >> 5;
  const int y    = blockIdx.x % kH;
  const int b    = blockIdx.x / kH;
  const float* dimg = depth + (size_t)b * (kH * kW);

  // 1) Zero all of LDS (halos, pad row; data region overwritten by async).
  for (int i = tid; i < kRows * kPitch; i += 256) smem[i] = 0.f;
  __syncthreads();

  // 2) Async-stage valid rows: 3 rows x 96 x b128 (16B per lane transfer).
  for (int i = tid; i < 3 * 96; i += 256) {
    const int r  = i / 96;
    const int c  = i - r * 96;
    const int yy = y + r - 1;
    if (yy >= 0 && yy < kH) {
      const float* g = dimg + (size_t)yy * kW + c * 4;
      unsigned ldsoff = (unsigned)(uintptr_t)&smem[r * kPitch + 4 + c * 4];
      asm volatile("global_load_async_to_lds_b128 %0, %1, off"
                   :: "v"(ldsoff), "v"(g) : "memory");
    }
  }
  asm volatile("s_wait_asynccnt 0" ::: "memory");
  __syncthreads();

  // 3) Per-lane A fragments (weights) + per-slot LDS base pointers.
  const int n    = lane & 15;   // A: row m (channel); B: column (pixel)
  const int half = lane >> 4;
  v2f A[3];
  const float* bp[3][2];
#pragma unroll
  for (int j = 0; j < 3; ++j) {
#pragma unroll
    for (int v = 0; v < 2; ++v) {
      const int t = 4 * j + 2 * half + v;   // tap 0..11 (>=9 -> pad row 3)
      float aw = 0.f;
      int row = 3, col = 4 + n;
      if (t < 9) {
        row = t / 3;                        // dy=-1,0,1 -> LDS row 0,1,2
        col = 4 + n + (t % 3 - 1);          // dx
        if (n < 4) aw = ws[n * 9 + t];      // zero rows m=4..15
      }
      A[j][v]  = aw;
      bp[j][v] = &smem[row * kPitch + col + wid * 48];
    }
  }

  // Uniform epilogue constants (scalar loads).
  const float sh0 = ws[36], sh1 = ws[37], sh2 = ws[38], sh3 = ws[39];
  const float w20 = ws[40], w21 = ws[41], w22 = ws[42], w23 = ws[43];
  const float sd = ws[44], bias = ws[45];

  const size_t outBase = ((size_t)b * kH + y) * kW + wid * 48;

  // 4) 3 segments of 16 px: 6 ds_load (imm offset) + 3 WMMA + epilogue each.
#pragma unroll
  for (int s = 0; s < 3; ++s) {
    v2f Bv[3];
#pragma unroll
    for (int j = 0; j < 3; ++j) {
      Bv[j][0] = bp[j][0][s * 16];
      Bv[j][1] = bp[j][1][s * 16];
    }
    v8f acc = {};
    acc = __builtin_amdgcn_wmma_f32_16x16x4_f32(false, A[0], false, Bv[0],
                                                (short)0, acc, false, false);
    acc = __builtin_amdgcn_wmma_f32_16x16x4_f32(false, A[1], false, Bv[1],
                                                (short)0, acc, false, false);
    acc = __builtin_amdgcn_wmma_f32_16x16x4_f32(false, A[2], false, Bv[2],
                                                (short)0, acc, false, false);

    // Lanes 0..15: Bv[1].x is tap 4 (dy=0,dx=0) = depth at this pixel.
    const float center = Bv[1][0];
    float o = fmaf(sd, center, bias);
    o = fmaf(w20, fmaxf(acc[0] + sh0, 0.f), o);
    o = fmaf(w21, fmaxf(acc[1] + sh1, 0.f), o);
    o = fmaf(w22, fmaxf(acc[2] + sh2, 0.f), o);
    o = fmaf(w23, fmaxf(acc[3] + sh3, 0.f), o);

    if (lane < 16) out[outBase + s * 16 + n] = o;  // D: channel r -> acc[r]
  }
}

extern "C" void kernel_launch(void* const* d_in, const int* in_sizes, int n_in,
                              void* d_out, int out_size, void* d_ws, size_t ws_size,
                              hipStream_t stream) {
  (void)in_sizes; (void)n_in; (void)out_size; (void)ws_size;
  // setup_inputs order: ae, depth, conv1_w, bn1_gamma, bn1_beta, bn1_mean,
  //                     bn1_var, conv2_w, conv2_b   (ae mathematically unused)
  const float* depth = (const float*)d_in[1];
  const float* w1    = (const float*)d_in[2];
  const float* gamma = (const float*)d_in[3];
  const float* beta  = (const float*)d_in[4];
  const float* mean  = (const float*)d_in[5];
  const float* var   = (const float*)d_in[6];
  const float* w2    = (const float*)d_in[7];
  const float* b2    = (const float*)d_in[8];
  float* out = (float*)d_out;
  float* ws  = (float*)d_ws;

  prep_kernel<<<1, 32, 0, stream>>>(w1, gamma, beta, mean, var, w2, b2, ws);

  const int blocks = kB * kH;   // one row per block: 12288 blocks x 256 threads
  tnet_wmma_kernel<<<blocks, 256, 0, stream>>>(depth, ws, out);
}